// SphereCloud_76828374991335
// MI455X (gfx1250) — compile-verified
//
#include <hip/hip_runtime.h>
#include <hip/hip_bf16.h>

typedef __attribute__((ext_vector_type(2))) float v2f;
typedef __attribute__((ext_vector_type(8))) float v8f;

#define EPS_R    1e-8f
#define T_MAX_R  100.0f
#define NRAYS    32768      // 16*64*32
#define NSPH     1024
#define SPH_TILES (NSPH / 16)

// One wave32 handles a tile of 16 rays vs all 1024 spheres (64 tiles of 16).
// A-matrix (16x4 f32): row M = ray, K = (x,y,z,const). Two A's share one B:
//   A_o = (ox,oy,oz,1), A_d = (dx,dy,dz,0), B col N = (-2cx,-2cy,-2cz, |c|^2-r^2)
//   accC[M][N] = -2 dot(o,c) + |c|^2 - r^2   -> c = |o|^2 + accC
//   accB[M][N] = -2 dot(d,c)                 -> b = 2 dot(d,o) + accB
__global__ __launch_bounds__(256)
void SphereCloud_wmma_kernel(const float* __restrict__ rays,
                             const float* __restrict__ centers,
                             const float* __restrict__ radii,
                             float* __restrict__ out)
{
    const int lane = threadIdx.x & 31;
    const int wave = threadIdx.x >> 5;
    const int half = lane >> 4;          // 0: K=0,1   1: K=2,3
    const int l15  = lane & 15;
    const int rayBase = (blockIdx.x * 8 + wave) * 16;   // 16-ray tile

    // ---- A matrices (16x4 f32 layout per ISA 7.12.2), branchless half-select ----
    const float* rp = rays + (size_t)(rayBase + l15) * 6;
    float ox = rp[0], oy = rp[1], oz = rp[2];
    float dx = rp[3], dy = rp[4], dz = rp[5];
    v2f a_o, a_d;
    a_o.x = half ? oz : ox;   a_o.y = half ? 1.0f : oy;
    a_d.x = half ? dz : dx;   a_d.y = half ? 0.0f : dy;

    // ---- per-lane epilogue rows: C layout => lane owns rows M = vr + 8*half ----
    float pa[8], p2do[8], poo[8], pinv[8];
#pragma unroll
    for (int vr = 0; vr < 8; ++vr) {
        const float* q = rays + (size_t)(rayBase + vr + 8 * half) * 6;
        float qox = q[0], qoy = q[1], qoz = q[2];
        float qdx = q[3], qdy = q[4], qdz = q[5];
        float a   = qdx * qdx + qdy * qdy + qdz * qdz;
        pa[vr]    = a;
        p2do[vr]  = 2.0f * (qdx * qox + qdy * qoy + qdz * qoz);
        poo[vr]   = qox * qox + qoy * qoy + qoz * qoz;
        pinv[vr]  = 1.0f / (2.0f * a);
    }

    float bestT[8];
    int   bestJ[8];
#pragma unroll
    for (int vr = 0; vr < 8; ++vr) { bestT[vr] = T_MAX_R; bestJ[vr] = -1; }

    // ---- sweep sphere tiles: fully branchless body ----
#pragma unroll 2
    for (int st = 0; st < SPH_TILES; ++st) {
        const int s = st * 16 + l15;      // this lane's sphere column (both halves)
        if (st + 1 < SPH_TILES)           // warm next tile (global_prefetch_b8)
            __builtin_prefetch(centers + (size_t)(s + 16) * 3, 0, 1);

        const float* cp = centers + (size_t)s * 3;
        float cx = cp[0], cy = cp[1], cz = cp[2];
        float rr = radii[s];
        float cc = cx * cx + cy * cy + cz * cz;

        v2f vb;                            // cndmask, no exec-mask churn
        vb.x = half ? (-2.0f * cz) : (-2.0f * cx);
        vb.y = half ? (cc - rr * rr) : (-2.0f * cy);

        v8f accC = {};
        v8f accB = {};
        accC = __builtin_amdgcn_wmma_f32_16x16x4_f32(
            false, a_o, false, vb, (short)0, accC, false, false);
        accB = __builtin_amdgcn_wmma_f32_16x16x4_f32(
            false, a_d, false, vb, (short)0, accB, false, false);

#pragma unroll
        for (int vr = 0; vr < 8; ++vr) {
            float b  = p2do[vr] + accB[vr];
            float c  = poo[vr]  + accC[vr];
            float d  = fmaf(b, b, -4.0f * pa[vr] * c);
            // raw v_sqrt_f32; d<=0 gives NaN which the compares reject below
            float ds = __builtin_amdgcn_sqrtf(d);
            float r1 = (-b + ds) * pinv[vr];
            float r2 = (-b - ds) * pinv[vr];
            r1 = (r1 >= EPS_R) ? r1 : T_MAX_R;   // NaN -> T_MAX
            r2 = (r2 >= EPS_R) ? r2 : T_MAX_R;
            float t  = fminf(r1, r2);
            bool hit = (d > 0.0f) & (t < bestT[vr]);
            bestT[vr] = hit ? t : bestT[vr];
            bestJ[vr] = hit ? s : bestJ[vr];
        }
    }

    // ---- reduce across the 16 lanes sharing each row; write outputs ----
#pragma unroll
    for (int vr = 0; vr < 8; ++vr) {
        float t = bestT[vr];
        int   j = bestJ[vr];
#pragma unroll
        for (int m = 1; m < 16; m <<= 1) {   // stays within 16-lane half
            float ot = __shfl_xor(t, m, 32);
            int   oj = __shfl_xor(j, m, 32);
            bool  take = (ot < t) | ((ot == t) & (oj < j));
            t = take ? ot : t;
            j = take ? oj : j;
        }
        if (l15 == vr) {                     // one writer lane per row
            const int ray = rayBase + vr + 8 * half;
            const float* q = rays + (size_t)ray * 6;
            float qox = q[0], qoy = q[1], qoz = q[2];
            float qdx = q[3], qdy = q[4], qdz = q[5];

            const bool active = (j >= 0);
            const float bt = t;              // stays T_MAX when no hit
            float px = qox + bt * qdx;
            float py = qoy + bt * qdy;
            float pz = qoz + bt * qdz;

            float nx = 0.0f, ny = 0.0f, nz = 0.0f;
            if (active) {
                // reference stores chunk-local index (chunks of 256)
                const int idx = j & 255;
                const float* cb = centers + (size_t)idx * 3;
                float fx = px - cb[0], fy = py - cb[1], fz = pz - cb[2];
                float nrm = fmaxf(sqrtf(fx * fx + fy * fy + fz * fz), 1e-12f);
                float inv = 1.0f / nrm;
                nx = fx * inv; ny = fy * inv; nz = fz * inv;
                px += nx * 1e-5f; py += ny * 1e-5f; pz += nz * 1e-5f;
            }
            out[(size_t)ray * 3 + 0] = px;
            out[(size_t)ray * 3 + 1] = py;
            out[(size_t)ray * 3 + 2] = pz;
            out[(size_t)NRAYS * 3 + (size_t)ray * 3 + 0] = nx;
            out[(size_t)NRAYS * 3 + (size_t)ray * 3 + 1] = ny;
            out[(size_t)NRAYS * 3 + (size_t)ray * 3 + 2] = nz;
            out[(size_t)NRAYS * 6 + ray] = bt;
            out[(size_t)NRAYS * 7 + ray] = active ? 1.0f : 0.0f;
        }
    }
}

extern "C" void kernel_launch(void* const* d_in, const int* in_sizes, int n_in,
                              void* d_out, int out_size, void* d_ws, size_t ws_size,
                              hipStream_t stream) {
    const float* rays    = (const float*)d_in[0];   // 32768 * 6
    const float* centers = (const float*)d_in[1];   // 1024 * 3
    const float* radii   = (const float*)d_in[2];   // 1024
    float* out = (float*)d_out;                     // p | n | best_dists | active

    // 2048 ray-tiles of 16 rays; 8 waves (tiles) per 256-thread block.
    dim3 grid(NRAYS / 16 / 8);
    dim3 block(256);
    SphereCloud_wmma_kernel<<<grid, block, 0, stream>>>(rays, centers, radii, out);
}